// CombinedLoss_16166256902865
// MI455X (gfx1250) — compile-verified
//
#include <hip/hip_runtime.h>
#include <math.h>

#define LAMBDA_C 10.0f
#define LAMBDA_E 10.0f
#define LAMBDA_O 10.0f
#define H_CONST  0.1f
#define EPS_CONST 1e-8f
#define INF_BITS 0x7f800000

typedef float v2f __attribute__((ext_vector_type(2)));
typedef float v8f __attribute__((ext_vector_type(8)));

// ---------------- init: set min-arrays to +inf bits ----------------
__global__ void init_min_kernel(int* rowc, int nrc, int* colc, int ncc,
                                int* rows, int nrs, int* cols, int ncs) {
  int i = blockIdx.x * blockDim.x + threadIdx.x;
  if (i < nrc) rowc[i] = INF_BITS;
  if (i < ncc) colc[i] = INF_BITS;
  if (i < nrs) rows[i] = INF_BITS;
  if (i < ncs) cols[i] = INF_BITS;
}

// ---------------- vertex squared norms ----------------
__global__ void vnorm_kernel(const float* __restrict__ pv, int N,
                             const float* __restrict__ tv, int M,
                             float* __restrict__ pnorm, float* __restrict__ tnorm) {
  int i = blockIdx.x * blockDim.x + threadIdx.x;
  if (i < M) {
    float x = tv[3*i], y = tv[3*i+1], z = tv[3*i+2];
    tnorm[i] = x*x + y*y + z*z;
  }
  if (i < N) {
    float x = pv[3*i], y = pv[3*i+1], z = pv[3*i+2];
    pnorm[i] = x*x + y*y + z*z;
  }
}

// ---------------- target face barycenters ----------------
__global__ void target_face_kernel(const float* __restrict__ tv, const int* __restrict__ tf,
                                   int Ft, float* __restrict__ bt, float* __restrict__ btnorm) {
  int i = blockIdx.x * blockDim.x + threadIdx.x;
  if (i >= Ft) return;
  int i0 = tf[3*i], i1 = tf[3*i+1], i2 = tf[3*i+2];
  const float third = 1.0f / 3.0f;
  float x = (tv[3*i0]   + tv[3*i1]   + tv[3*i2]  ) * third;
  float y = (tv[3*i0+1] + tv[3*i1+1] + tv[3*i2+1]) * third;
  float z = (tv[3*i0+2] + tv[3*i1+2] + tv[3*i2+2]) * third;
  bt[3*i] = x; bt[3*i+1] = y; bt[3*i+2] = z;
  btnorm[i] = x*x + y*y + z*z;
}

// ---------------- pred face: tri verts, barycenter, normal, plane offset ----------------
__global__ void pred_face_kernel(const float* __restrict__ pv, const int* __restrict__ pf,
                                 int F, float* __restrict__ tri, float* __restrict__ bp,
                                 float* __restrict__ bpnorm, float* __restrict__ nhat,
                                 float* __restrict__ d0) {
  int i = blockIdx.x * blockDim.x + threadIdx.x;
  if (i >= F) return;
  int i0 = pf[3*i], i1 = pf[3*i+1], i2 = pf[3*i+2];
  float v0x = pv[3*i0], v0y = pv[3*i0+1], v0z = pv[3*i0+2];
  float v1x = pv[3*i1], v1y = pv[3*i1+1], v1z = pv[3*i1+2];
  float v2x = pv[3*i2], v2y = pv[3*i2+1], v2z = pv[3*i2+2];
  tri[9*i+0] = v0x; tri[9*i+1] = v0y; tri[9*i+2] = v0z;
  tri[9*i+3] = v1x; tri[9*i+4] = v1y; tri[9*i+5] = v1z;
  tri[9*i+6] = v2x; tri[9*i+7] = v2y; tri[9*i+8] = v2z;
  const float third = 1.0f / 3.0f;
  float bx = (v0x + v1x + v2x) * third;
  float by = (v0y + v1y + v2y) * third;
  float bz = (v0z + v1z + v2z) * third;
  bp[3*i] = bx; bp[3*i+1] = by; bp[3*i+2] = bz;
  bpnorm[i] = bx*bx + by*by + bz*bz;
  float e1x = v1x - v0x, e1y = v1y - v0y, e1z = v1z - v0z;
  float e2x = v2x - v0x, e2y = v2y - v0y, e2z = v2z - v0z;
  float nx = e1y*e2z - e1z*e2y;
  float ny = e1z*e2x - e1x*e2z;
  float nz = e1x*e2y - e1y*e2x;
  float inv = 1.0f / (sqrtf(nx*nx + ny*ny + nz*nz) + EPS_CONST);
  nx *= inv; ny *= inv; nz *= inv;
  nhat[3*i] = nx; nhat[3*i+1] = ny; nhat[3*i+2] = nz;
  d0[i] = nx*v0x + ny*v0y + nz*v0z;
}

// ---------------- pairwise squared-distance min via WMMA f32 16x16x4 ----------------
// d2[m][n] = |A_m|^2 + |B_n|^2 - 2 A_m . B_n ; cross term via V_WMMA_F32_16X16X4_F32,
// K=3 padded to 4, factor -2 folded into A. Each wave: 128 columns x (nA/16/gridDim.y)
// row-tiles. Row-min: shfl_xor over the 16-lane half; col-min: shfl_xor(16).
// Combined across waves with exact int-bit atomicMin (d2 clamped >= 0).
__global__ __launch_bounds__(256)
void pairwise_min_wmma(const float* __restrict__ A, const float* __restrict__ An, int nA,
                       const float* __restrict__ B, const float* __restrict__ Bn, int nB,
                       int* __restrict__ rowmin, int* __restrict__ colmin) {
  const int lane = threadIdx.x & 31;
  const int ln   = lane & 15;
  const int half = lane >> 4;
  const int wave = (blockIdx.x * blockDim.x + threadIdx.x) >> 5;
  const int colBase = wave * 128;
  if (colBase >= nB) return;
  const int rowTiles = nA >> 4;
  const int rtPer = rowTiles / gridDim.y;
  const int rt0 = blockIdx.y * rtPer;
  const float INF = __int_as_float(INF_BITS);

  float b0[8], b1[8], tn[8], cmin[8];
#pragma unroll
  for (int ct = 0; ct < 8; ++ct) {
    int j = colBase + ct * 16 + ln;
    float x = B[3*j], y = B[3*j+1], z = B[3*j+2];
    b0[ct] = half ? z : x;        // B layout: lanes<16 carry K=0,1 ; lanes>=16 carry K=2,3
    b1[ct] = half ? 0.0f : y;
    tn[ct] = Bn[j];
    cmin[ct] = INF;
  }

  for (int rt = rt0; rt < rt0 + rtPer; ++rt) {
    int i = rt * 16 + ln;
    float ax = A[3*i], ay = A[3*i+1], az = A[3*i+2];
    v2f av;
    av.x = half ? (-2.0f * az) : (-2.0f * ax);   // A 16x4: lanes<16 K=0,1 ; lanes>=16 K=2,3
    av.y = half ? 0.0f         : (-2.0f * ay);
    float pn[8], rmin[8];
#pragma unroll
    for (int v = 0; v < 8; ++v) {
      pn[v] = An[rt * 16 + 8 * half + v];  // C/D layout: M = v + 8*(lane/16)
      rmin[v] = INF;
    }
#pragma unroll
    for (int ct = 0; ct < 8; ++ct) {
      v2f bv; bv.x = b0[ct]; bv.y = b1[ct];
      v8f acc = {};
      acc = __builtin_amdgcn_wmma_f32_16x16x4_f32(false, av, false, bv,
                                                  (short)0, acc, false, false);
#pragma unroll
      for (int v = 0; v < 8; ++v) {
        float d2 = fmaxf(acc[v] + pn[v] + tn[ct], 0.0f);
        rmin[v] = fminf(rmin[v], d2);
        cmin[ct] = fminf(cmin[ct], d2);
      }
    }
#pragma unroll
    for (int v = 0; v < 8; ++v) {
      float m = rmin[v];
      m = fminf(m, __shfl_xor(m, 1, 32));
      m = fminf(m, __shfl_xor(m, 2, 32));
      m = fminf(m, __shfl_xor(m, 4, 32));
      m = fminf(m, __shfl_xor(m, 8, 32));
      if (ln == 0) atomicMin(&rowmin[rt * 16 + 8 * half + v], __float_as_int(m));
    }
  }
#pragma unroll
  for (int ct = 0; ct < 8; ++ct) {
    float m = fminf(cmin[ct], __shfl_xor(cmin[ct], 16, 32));
    if (half == 0) atomicMin(&colmin[colBase + ct * 16 + ln], __float_as_int(m));
  }
}

// ---------------- fused F x F pair terms: collision + edge-crossing + overlap ----------------
__global__ __launch_bounds__(256)
void pair_terms_kernel(const float* __restrict__ tri, const float* __restrict__ bp,
                       const float* __restrict__ nhat, const float* __restrict__ d0,
                       const float* __restrict__ probs, int F,
                       float* __restrict__ partials) {
  int idx = blockIdx.x * blockDim.x + threadIdx.x;
  int i = idx / F, j = idx - i * F;
  float val = 0.0f;
  if (i < F && i != j) {
    float bix = bp[3*i], biy = bp[3*i+1], biz = bp[3*i+2];
    float bjx = bp[3*j], bjy = bp[3*j+1], bjz = bp[3*j+2];
    float dx = bix - bjx, dy = biy - bjy, dz = biz - bjz;
    float cd2 = dx*dx + dy*dy + dz*dz;
    float gate = expf(-cd2 * (1.0f / H_CONST));
    float nx = nhat[3*i], ny = nhat[3*i+1], nz = nhat[3*i+2];
    float di = d0[i];
    float tj[9], ti[9];
#pragma unroll
    for (int k = 0; k < 9; ++k) { tj[k] = tri[9*j + k]; ti[k] = tri[9*i + k]; }
    // collision: plane of i vs vertices of j
    float s0 = nx*tj[0] + ny*tj[1] + nz*tj[2] - di;
    float s1 = nx*tj[3] + ny*tj[4] + nz*tj[5] - di;
    float s2 = nx*tj[6] + ny*tj[7] + nz*tj[8] - di;
    float smax = fmaxf(fmaxf(s0, s1), s2);
    float smin = fminf(fminf(s0, s1), s2);
    float pen_col = fmaxf(0.0f, -(smax * smin));
    // overlap: plane of i vs barycenter of j
    float dp = fabsf(nx*bjx + ny*bjy + nz*bjz - di);
    float pen_ov = fmaxf(0.0f, H_CONST - dp);
    // edge-crossing: 3x3 segment-segment distances
    float pen_edge = 0.0f;
#pragma unroll
    for (int ea = 0; ea < 3; ++ea) {
      int a0 = 3 * ea, a1 = 3 * ((ea + 1) % 3);
      float p1x = ti[a0], p1y = ti[a0+1], p1z = ti[a0+2];
      float d1x = ti[a1] - p1x, d1y = ti[a1+1] - p1y, d1z = ti[a1+2] - p1z;
      float aa = d1x*d1x + d1y*d1y + d1z*d1z;
#pragma unroll
      for (int eb = 0; eb < 3; ++eb) {
        int c0 = 3 * eb, c1 = 3 * ((eb + 1) % 3);
        float p2x = tj[c0], p2y = tj[c0+1], p2z = tj[c0+2];
        float d2x = tj[c1] - p2x, d2y = tj[c1+1] - p2y, d2z = tj[c1+2] - p2z;
        float ee = d2x*d2x + d2y*d2y + d2z*d2z;
        float rx = p1x - p2x, ry = p1y - p2y, rz = p1z - p2z;
        float ff = d2x*rx + d2y*ry + d2z*rz;
        float cc = d1x*rx + d1y*ry + d1z*rz;
        float bb = d1x*d2x + d1y*d2y + d1z*d2z;
        float denom = aa * ee - bb * bb;
        float ss = (bb * ff - cc * ee) / (denom + EPS_CONST);
        ss = fminf(fmaxf(ss, 0.0f), 1.0f);
        float tt = (bb * ss + ff) / (ee + EPS_CONST);
        tt = fminf(fmaxf(tt, 0.0f), 1.0f);
        ss = (bb * tt - cc) / (aa + EPS_CONST);
        ss = fminf(fmaxf(ss, 0.0f), 1.0f);
        float cx = (p1x + ss * d1x) - (p2x + tt * d2x);
        float cy = (p1y + ss * d1y) - (p2y + tt * d2y);
        float cz = (p1z + ss * d1z) - (p2z + tt * d2z);
        float dist = sqrtf(cx*cx + cy*cy + cz*cz + EPS_CONST);
        pen_edge += fmaxf(0.0f, H_CONST - dist);
      }
    }
    val = probs[i] * gate * (LAMBDA_C * pen_col + LAMBDA_E * pen_edge + LAMBDA_O * pen_ov);
  }
  __shared__ float sd[256];
  sd[threadIdx.x] = val;
  __syncthreads();
  for (int s = 128; s > 0; s >>= 1) {
    if (threadIdx.x < s) sd[threadIdx.x] += sd[threadIdx.x + s];
    __syncthreads();
  }
  if (threadIdx.x == 0) partials[blockIdx.x] = sd[0];
}

// ---------------- deterministic final reduction ----------------
__global__ __launch_bounds__(256)
void final_reduce_kernel(const int* __restrict__ rowc, int N, const int* __restrict__ colc, int M,
                         const int* __restrict__ rows, const int* __restrict__ cols,
                         const float* __restrict__ probs, int F, int Ft,
                         const float* __restrict__ partials, int nPart,
                         float* __restrict__ out) {
  int tid = threadIdx.x;
  double acc = 0.0;
  for (int i = tid; i < N;  i += 256) acc += (double)__int_as_float(rowc[i]) / (double)N;
  for (int i = tid; i < M;  i += 256) acc += (double)__int_as_float(colc[i]) / (double)M;
  for (int i = tid; i < F;  i += 256) acc += (double)(probs[i] * __int_as_float(rows[i])) / (double)F;
  for (int i = tid; i < Ft; i += 256) acc += (double)__int_as_float(cols[i]) / (double)Ft;
  for (int i = tid; i < nPart; i += 256) acc += (double)partials[i] / (double)F;
  __shared__ double sd[256];
  sd[tid] = acc;
  __syncthreads();
  for (int s = 128; s > 0; s >>= 1) {
    if (tid < s) sd[tid] += sd[tid + s];
    __syncthreads();
  }
  if (tid == 0) out[0] = (float)sd[0];
}

static inline int imax2(int a, int b) { return a > b ? a : b; }

extern "C" void kernel_launch(void* const* d_in, const int* in_sizes, int n_in,
                              void* d_out, int out_size, void* d_ws, size_t ws_size,
                              hipStream_t stream) {
  const float* pv    = (const float*)d_in[0];
  const float* probs = (const float*)d_in[1];
  const float* tv    = (const float*)d_in[2];
  const int*   pf    = (const int*)d_in[3];
  const int*   tf    = (const int*)d_in[4];
  const int N  = in_sizes[0] / 3;   // 512
  const int F  = in_sizes[1];       // 1024
  const int M  = in_sizes[2] / 3;   // 32768
  const int Ft = in_sizes[4] / 3;   // 65536
  (void)n_in; (void)out_size; (void)ws_size;

  float* w = (float*)d_ws;
  float* tnorm  = w; w += M;
  float* pnorm  = w; w += N;
  float* bt     = w; w += 3 * Ft;
  float* btnorm = w; w += Ft;
  float* tri    = w; w += 9 * F;
  float* bp     = w; w += 3 * F;
  float* bpnorm = w; w += F;
  float* nhat   = w; w += 3 * F;
  float* d0     = w; w += F;
  int* rowc = (int*)w; w += N;
  int* colc = (int*)w; w += M;
  int* rows = (int*)w; w += F;
  int* cols = (int*)w; w += Ft;
  float* partials = w;
  const int nPairBlocks = (F * F) / 256;

  int maxInit = imax2(imax2(N, M), imax2(F, Ft));
  init_min_kernel<<<(maxInit + 255) / 256, 256, 0, stream>>>(rowc, N, colc, M, rows, F, cols, Ft);
  vnorm_kernel<<<(imax2(N, M) + 255) / 256, 256, 0, stream>>>(pv, N, tv, M, pnorm, tnorm);
  target_face_kernel<<<(Ft + 255) / 256, 256, 0, stream>>>(tv, tf, Ft, bt, btnorm);
  pred_face_kernel<<<(F + 255) / 256, 256, 0, stream>>>(pv, pf, F, tri, bp, bpnorm, nhat, d0);

  {
    // chamfer: 512 x 32768. waves = M/128; 8 waves/block; 4 row-groups of 8 tiles.
    dim3 grid(M / 128 / 8, 4);
    pairwise_min_wmma<<<grid, 256, 0, stream>>>(pv, pnorm, N, tv, tnorm, M, rowc, colc);
  }
  {
    // surface: 1024 x 65536. waves = Ft/128; 8 row-groups of 8 tiles.
    dim3 grid(Ft / 128 / 8, 8);
    pairwise_min_wmma<<<grid, 256, 0, stream>>>(bp, bpnorm, F, bt, btnorm, Ft, rows, cols);
  }
  pair_terms_kernel<<<nPairBlocks, 256, 0, stream>>>(tri, bp, nhat, d0, probs, F, partials);
  final_reduce_kernel<<<1, 256, 0, stream>>>(rowc, N, colc, M, rows, cols,
                                             probs, F, Ft, partials, nPairBlocks,
                                             (float*)d_out);
}